// CenterLoss_76209899700452
// MI455X (gfx1250) — compile-verified
//
#include <hip/hip_runtime.h>
#include <hip/hip_bf16.h>

// CenterLoss on MI455X (gfx1250, wave32).
// Only the label-gathered diagonal of the distance matrix is needed:
//   out = mean_i ||x_i - centers[labels[i]]||^2
// Memory-bound (~16.8 MB @ 23.3 TB/s ~ 0.7us). The per-row sum-of-squares
// reduction is done on the matrix pipe via V_WMMA_F32_16X16X4_F32 with an
// all-ones B matrix: D[m,n] = sum_k A[m,k], A[m,k] = (x-c)^2.

typedef float v2f __attribute__((ext_vector_type(2)));
typedef float v8f __attribute__((ext_vector_type(8)));

#define CL_BATCH 4096
#define CL_DIM   512
#define CL_TILES (CL_BATCH / 16)   // 256 tiles of 16 rows
#define CL_KSLC  4                 // 4 K-slices of 128 each
#define CL_WAVES (CL_TILES * CL_KSLC)  // 1024 waves total

__global__ __launch_bounds__(256)
void center_loss_partial_kernel(const float* __restrict__ x,
                                const float* __restrict__ centers,
                                const int*   __restrict__ labels,
                                float*       __restrict__ partials) {
    const int lane = threadIdx.x & 31;
    const int wave = threadIdx.x >> 5;
    const int gid  = blockIdx.x * 8 + wave;   // 0..1023
    const int tile  = gid >> 2;               // 0..255 : 16 batch rows
    const int slice = gid & 3;                // 0..3   : 128 k-values

    // A-matrix (16x4 f32) layout: lanes 0-15 -> M=lane, K = k0+{0,1};
    // lanes 16-31 -> M=lane-16, K = k0+{2,3}. Each lane loads a float2.
    const int m     = lane & 15;
    const int row   = tile * 16 + m;
    const int kbase = slice * 128 + ((lane >> 4) << 1);

    const float* __restrict__ xr = x + (size_t)row * CL_DIM + kbase;
    const int    lbl             = labels[row];
    const float* __restrict__ cr = centers + (size_t)lbl * CL_DIM + kbase;

    v2f ones; ones[0] = 1.0f; ones[1] = 1.0f;   // B = 4x16 all-ones
    v8f acc = {};                               // C/D accumulator (16x16 f32)

    #pragma unroll 8
    for (int k = 0; k < 128; k += 4) {
        v2f xv = *(const v2f*)(xr + k);
        v2f cv = *(const v2f*)(cr + k);
        float d0 = xv[0] - cv[0];
        float d1 = xv[1] - cv[1];
        v2f a;
        a[0] = d0 * d0;
        a[1] = d1 * d1;
        // D = A x B + C : every column n of D gets the row sum of A.
        acc = __builtin_amdgcn_wmma_f32_16x16x4_f32(
            /*neg_a=*/false, a, /*neg_b=*/false, ones,
            /*c_mod=*/(short)0, acc, /*reuse_a=*/false, /*reuse_b=*/false);
    }

    // C/D layout: lanes 0-15 hold D[r, lane] in acc[r] (rows 0..7),
    // lanes 16-31 hold rows 8..15. Columns are identical (B = ones), so
    // lane 0 carries sum(d_0..d_7) and lane 16 carries sum(d_8..d_15).
    float s = acc[0] + acc[1] + acc[2] + acc[3] +
              acc[4] + acc[5] + acc[6] + acc[7];

    if ((lane & 15) == 0) {
        partials[gid * 2 + (lane >> 4)] = s;   // 2048 partials total
    }
}

__global__ __launch_bounds__(256)
void center_loss_reduce_kernel(const float* __restrict__ partials,
                               float* __restrict__ out) {
    __shared__ float sm[256];
    const int t = threadIdx.x;
    float s = 0.0f;
    #pragma unroll
    for (int i = 0; i < 2 * CL_WAVES / 256; ++i) {
        s += partials[t + 256 * i];
    }
    sm[t] = s;
    __syncthreads();
    for (int off = 128; off > 0; off >>= 1) {
        if (t < off) sm[t] += sm[t + off];
        __syncthreads();
    }
    if (t == 0) out[0] = sm[0] * (1.0f / (float)CL_BATCH);
}

extern "C" void kernel_launch(void* const* d_in, const int* in_sizes, int n_in,
                              void* d_out, int out_size, void* d_ws, size_t ws_size,
                              hipStream_t stream) {
    const float* x       = (const float*)d_in[0];   // [4096, 512] f32
    const float* centers = (const float*)d_in[1];   // [7001, 512] f32
    const int*   labels  = (const int*)d_in[2];     // [4096] i32
    float*       out     = (float*)d_out;           // [1] f32
    float*       partials = (float*)d_ws;           // 2048 f32 = 8 KB scratch

    // 1024 waves = 128 blocks x 8 waves(256 thr); each wave: 16 rows x 128 k.
    center_loss_partial_kernel<<<CL_WAVES / 8, 256, 0, stream>>>(
        x, centers, labels, partials);
    center_loss_reduce_kernel<<<1, 256, 0, stream>>>(partials, out);
}